// GraphSAGENet_48679159333231
// MI455X (gfx1250) — compile-verified
//
#include <hip/hip_runtime.h>
#include <hip/hip_bf16.h>

// GraphSAGE (2x SAGEConv, mean aggr) for MI455X / gfx1250, wave32.
//
// Pipeline (all on `stream`):
//   zero_kernel     : clear cnt/agg1/aggg accumulators in ws
//   pad_w_kernel    : pad weights to 16x16 tiles (branch-free WMMA fragments)
//   xpad_kernel     : pad x to N x 16 (64B rows -> b128 gathers, b64 fragments)
//   scatter1_kernel : per-edge atomicAdd of x[src] (14ch) + degree into agg1/cnt
//   lin1_kernel     : per 16-node tile (one wave32):
//                       h = relu(mean1 @ W1_l^T + b1 + x @ W1_r^T)   [WMMA f32 16x16x4]
//                       g = h @ W2_l^T  (LDS transpose D->A layout)  [WMMA f32 16x16x4]
//   scatter2_kernel : per-edge atomicAdd of g[src] (8ch) into aggg
//                     (mean(h)@W2_l^T == mean(h@W2_l^T) -> only 8ch of atomics)
//   lin2_kernel     : out = aggg/cnt + b2 + h @ W2_r^T               [WMMA f32 16x16x4]

typedef __attribute__((ext_vector_type(2))) float v2f;
typedef __attribute__((ext_vector_type(8))) float v8f;

#define IN_CH  14
#define HIDF   16
#define OUT_CH 8

__device__ __forceinline__ void atomAddF(float* p, float v) {
  __hip_atomic_fetch_add(p, v, __ATOMIC_RELAXED, __HIP_MEMORY_SCOPE_AGENT);
}

__global__ void __launch_bounds__(256) zero_kernel(float* p, long n4) {
  long i = (long)blockIdx.x * blockDim.x + threadIdx.x;
  long stride = (long)gridDim.x * blockDim.x;
  float4* p4 = (float4*)p;
  float4 z = make_float4(0.f, 0.f, 0.f, 0.f);
  for (; i < n4; i += stride) p4[i] = z;
}

// Pad all weight matrices into 16x16 f32 tiles + padded bias (single block).
// wpad layout (floats): [0,256) W1lp | [256,512) W1rp | [512,768) W2lp
//                       [768,1024) W2rp | [1024,1040) b2p
__global__ void __launch_bounds__(256) pad_w_kernel(
    const float* __restrict__ W1l, const float* __restrict__ W1r,
    const float* __restrict__ W2l, const float* __restrict__ W2r,
    const float* __restrict__ b2, float* __restrict__ wpad) {
  const int t = threadIdx.x;        // t = n*16 + k
  const int n = t >> 4, k = t & 15;
  wpad[t]        = (k < IN_CH) ? W1l[n * IN_CH + k] : 0.f;
  wpad[256 + t]  = (k < IN_CH) ? W1r[n * IN_CH + k] : 0.f;
  wpad[512 + t]  = (n < OUT_CH) ? W2l[n * HIDF + k] : 0.f;
  wpad[768 + t]  = (n < OUT_CH) ? W2r[n * HIDF + k] : 0.f;
  if (t < 16) wpad[1024 + t] = (t < OUT_CH) ? b2[t] : 0.f;
}

__global__ void __launch_bounds__(256) xpad_kernel(
    const float* __restrict__ x, float* __restrict__ xpad, int N) {
  int i = blockIdx.x * blockDim.x + threadIdx.x;   // over N*16
  if (i >= N * 16) return;
  const int row = i >> 4, k = i & 15;
  xpad[i] = (k < IN_CH) ? x[(long)row * IN_CH + k] : 0.f;
}

__global__ void __launch_bounds__(256) scatter1_kernel(
    const long long* __restrict__ ei, const float* __restrict__ xpad,
    float* __restrict__ agg1, float* __restrict__ cnt, int E) {
  int e = blockIdx.x * blockDim.x + threadIdx.x;
  if (e >= E) return;
  __builtin_prefetch(&ei[e + 8192], 0, 1);             // stream src indices
  __builtin_prefetch(&ei[(long)E + e + 8192], 0, 1);   // stream dst indices
  int src = (int)ei[e];
  int dst = (int)ei[(long)E + e];
  atomAddF(&cnt[dst], 1.0f);
  const float4* xs = (const float4*)(xpad + (long)src * 16);  // L2-resident
  float4 r0 = xs[0], r1 = xs[1], r2 = xs[2];
  float2 r3 = *(const float2*)(xpad + (long)src * 16 + 12);
  float* a = agg1 + (long)dst * 16;
  atomAddF(&a[0],  r0.x); atomAddF(&a[1],  r0.y);
  atomAddF(&a[2],  r0.z); atomAddF(&a[3],  r0.w);
  atomAddF(&a[4],  r1.x); atomAddF(&a[5],  r1.y);
  atomAddF(&a[6],  r1.z); atomAddF(&a[7],  r1.w);
  atomAddF(&a[8],  r2.x); atomAddF(&a[9],  r2.y);
  atomAddF(&a[10], r2.z); atomAddF(&a[11], r2.w);
  atomAddF(&a[12], r3.x); atomAddF(&a[13], r3.y);
}

__global__ void __launch_bounds__(256) scatter2_kernel(
    const long long* __restrict__ ei, const float* __restrict__ g,
    float* __restrict__ aggg, int E) {   // g/aggg rows stride 16; aggg cols 8..15 stay 0
  int e = blockIdx.x * blockDim.x + threadIdx.x;
  if (e >= E) return;
  __builtin_prefetch(&ei[e + 8192], 0, 1);
  __builtin_prefetch(&ei[(long)E + e + 8192], 0, 1);
  int src = (int)ei[e];
  int dst = (int)ei[(long)E + e];
  const float4* gs = (const float4*)(g + (long)src * 16);
  float4 g0 = gs[0], g1 = gs[1];
  float* a = aggg + (long)dst * 16;
  atomAddF(&a[0], g0.x); atomAddF(&a[1], g0.y);
  atomAddF(&a[2], g0.z); atomAddF(&a[3], g0.w);
  atomAddF(&a[4], g1.x); atomAddF(&a[5], g1.y);
  atomAddF(&a[6], g1.z); atomAddF(&a[7], g1.w);
}

// One wave32 per 16-node tile. WMMA f32 16x16x4 layouts (ISA 7.12.2):
//   A 16x4 : lanes 0-15 -> M=lane, {K=0,K=1}; lanes 16-31 -> M=lane-16, {K=2,K=3}
//   B 4x16 : lanes 0-15 -> N=lane, {K=0,K=1}; lanes 16-31 -> N=lane-16, {K=2,K=3}
//   C/D    : lane owns column N=lane&15; VGPR v holds row M = v + 8*(lane>>4)
__global__ void __launch_bounds__(256) lin1_kernel(
    const float* __restrict__ xpad, const float* __restrict__ agg1,
    const float* __restrict__ cnt, const float* __restrict__ wpad,
    const float* __restrict__ b1,
    float* __restrict__ h, float* __restrict__ g, int N) {
  __shared__ float lds_h[8][16][18];   // stride 18: 8B-aligned pairs, no bank dup
  const float* W1lp = wpad;
  const float* W1rp = wpad + 256;
  const float* W2lp = wpad + 512;
  const int lane  = threadIdx.x & 31;
  const int wslot = threadIdx.x >> 5;
  const int tile  = blockIdx.x * 8 + wslot;
  const int m     = lane & 15;
  const int hi    = lane >> 4;
  const int kgrp  = hi * 2;
  const int n     = m;                          // C/D column owned by this lane
  const int nodeA  = tile * 16 + m;
  const int nodeAc = nodeA < N ? nodeA : N - 1; // clamp loads; EXEC stays all-1
  const bool fullTile = (tile * 16 + 16) <= N;  // wave-uniform
  const float invA = 1.0f / fmaxf(cnt[nodeAc], 1.0f);
  const float* A1row = agg1 + (long)nodeAc * 16;  // cols 14,15 are zero
  const float* A2row = xpad + (long)nodeAc * 16;  // cols 14,15 are zero

  v8f acc;
  const float bias = b1[n];
#pragma unroll
  for (int v = 0; v < 8; ++v) acc[v] = bias;

  // mean1 @ W1_l^T   (K padded to 16 -> 4 steps of K=4, branch-free)
#pragma unroll
  for (int s = 0; s < 4; ++s) {
    const int k0 = 4 * s + kgrp;
    v2f a, b;
    a.x = A1row[k0] * invA;  a.y = A1row[k0 + 1] * invA;
    b.x = W1lp[n * 16 + k0]; b.y = W1lp[n * 16 + k0 + 1];
    acc = __builtin_amdgcn_wmma_f32_16x16x4_f32(false, a, false, b, (short)0,
                                                acc, false, false);
  }
  // + x @ W1_r^T
#pragma unroll
  for (int s = 0; s < 4; ++s) {
    const int k0 = 4 * s + kgrp;
    v2f a, b;
    a.x = A2row[k0];         a.y = A2row[k0 + 1];
    b.x = W1rp[n * 16 + k0]; b.y = W1rp[n * 16 + k0 + 1];
    acc = __builtin_amdgcn_wmma_f32_16x16x4_f32(false, a, false, b, (short)0,
                                                acc, false, false);
  }

  // ReLU; stage tile in LDS (D layout); store h (uniform fast path)
#pragma unroll
  for (int v = 0; v < 8; ++v) {
    acc[v] = fmaxf(acc[v], 0.f);
    lds_h[wslot][v + 8 * hi][n] = acc[v];
  }
  if (fullTile) {
#pragma unroll
    for (int v = 0; v < 8; ++v)
      h[(long)(tile * 16 + v + 8 * hi) * HIDF + n] = acc[v];
  } else {
#pragma unroll
    for (int v = 0; v < 8; ++v) {
      const int node = tile * 16 + v + 8 * hi;
      if (node < N) h[(long)node * HIDF + n] = acc[v];
    }
  }
  asm volatile("s_wait_dscnt 0" ::: "memory");  // wave-private LDS RAW fence

  // g = h @ W2_l^T   (pre-aggregation projection; halves layer-2 atomics)
  v8f accg = {};
#pragma unroll
  for (int s = 0; s < 4; ++s) {
    const int k0 = 4 * s + kgrp;
    v2f a, b;
    a.x = lds_h[wslot][m][k0];
    a.y = lds_h[wslot][m][k0 + 1];
    b.x = W2lp[n * 16 + k0]; b.y = W2lp[n * 16 + k0 + 1];
    accg = __builtin_amdgcn_wmma_f32_16x16x4_f32(false, a, false, b, (short)0,
                                                 accg, false, false);
  }
  // g rows stride 16; cols 8..15 hold junk that scatter2 never reads
  if (fullTile) {
#pragma unroll
    for (int v = 0; v < 8; ++v)
      g[(long)(tile * 16 + v + 8 * hi) * 16 + n] = accg[v];
  } else {
#pragma unroll
    for (int v = 0; v < 8; ++v) {
      const int node = tile * 16 + v + 8 * hi;
      if (node < N) g[(long)node * 16 + n] = accg[v];
    }
  }
}

__global__ void __launch_bounds__(256) lin2_kernel(
    const float* __restrict__ h, const float* __restrict__ aggg,
    const float* __restrict__ cnt, const float* __restrict__ wpad,
    float* __restrict__ out, int N) {
  const float* W2rp = wpad + 768;
  const float* b2p  = wpad + 1024;
  const int lane  = threadIdx.x & 31;
  const int wslot = threadIdx.x >> 5;
  const int tile  = blockIdx.x * 8 + wslot;
  const int m     = lane & 15;
  const int hi    = lane >> 4;
  const int kgrp  = hi * 2;
  const int n     = m;
  const int nodeA  = tile * 16 + m;
  const int nodeAc = nodeA < N ? nodeA : N - 1;
  const bool fullTile = (tile * 16 + 16) <= N;
  const float* Arow = h + (long)nodeAc * HIDF;

  // One precise division per lane (for its own row m); rows for the C init
  // are fetched cross-lane with ds_bpermute via __shfl.
  const float invA = 1.0f / fmaxf(cnt[nodeAc], 1.0f);

  // C init = mean(h @ W2_l^T) + b2  (aggg stride 16, cols 8..15 zero; b2p padded)
  v8f acc;
  const float bias = b2p[n];
#pragma unroll
  for (int v = 0; v < 8; ++v) {
    const int row = v + 8 * hi;
    const int node = tile * 16 + row;
    const int nodec = node < N ? node : N - 1;
    const float inv = __shfl(invA, row, 32);   // inv for row (held by lane==row)
    acc[v] = aggg[(long)nodec * 16 + n] * inv + bias;
  }
  // + h @ W2_r^T   (K=16 -> 4 steps)
#pragma unroll
  for (int s = 0; s < 4; ++s) {
    const int k0 = 4 * s + kgrp;
    v2f a, b;
    a.x = Arow[k0];          a.y = Arow[k0 + 1];
    b.x = W2rp[n * 16 + k0]; b.y = W2rp[n * 16 + k0 + 1];
    acc = __builtin_amdgcn_wmma_f32_16x16x4_f32(false, a, false, b, (short)0,
                                                acc, false, false);
  }
  if (fullTile) {
    if (n < OUT_CH) {          // single exec region for all 8 stores
#pragma unroll
      for (int v = 0; v < 8; ++v)
        out[(long)(tile * 16 + v + 8 * hi) * OUT_CH + n] = acc[v];
    }
  } else {
#pragma unroll
    for (int v = 0; v < 8; ++v) {
      const int node = tile * 16 + v + 8 * hi;
      if (node < N && n < OUT_CH) out[(long)node * OUT_CH + n] = acc[v];
    }
  }
}

extern "C" void kernel_launch(void* const* d_in, const int* in_sizes, int n_in,
                              void* d_out, int out_size, void* d_ws, size_t ws_size,
                              hipStream_t stream) {
  const float*     x   = (const float*)d_in[0];
  const long long* ei  = (const long long*)d_in[1];
  // d_in[2] = edge_attr: unused by the reference forward pass
  const float* W1l = (const float*)d_in[3];
  const float* b1  = (const float*)d_in[4];
  const float* W1r = (const float*)d_in[5];
  const float* W2l = (const float*)d_in[6];
  const float* b2  = (const float*)d_in[7];
  const float* W2r = (const float*)d_in[8];

  const int N = in_sizes[0] / IN_CH;
  const int E = in_sizes[1] / 2;

  // ws layout (floats):
  // [wpad 1280][cnt N][agg1 16N][aggg 16N][h 16N][g 16N][xpad 16N] = 1280 + 81N
  float* ws   = (float*)d_ws;
  float* wpad = ws;
  float* cnt  = ws + 1280;
  float* agg1 = cnt + (long)N;
  float* aggg = agg1 + (long)N * 16;
  float* h    = aggg + (long)N * 16;
  float* g    = h + (long)N * HIDF;
  float* xpad = g + (long)N * 16;

  // Zero cnt+agg1+aggg (33N floats; 4-float rounding spills into h, which is
  // fully overwritten by lin1 before any read).
  const long zero4 = ((long)N * 33 + 3) / 4;
  long zb = (zero4 + 255) / 256;
  if (zb > 4096) zb = 4096;
  if (zb < 1) zb = 1;
  zero_kernel<<<(int)zb, 256, 0, stream>>>(cnt, zero4);

  pad_w_kernel<<<1, 256, 0, stream>>>(W1l, W1r, W2l, W2r, b2, wpad);
  xpad_kernel<<<(N * 16 + 255) / 256, 256, 0, stream>>>(x, xpad, N);

  scatter1_kernel<<<(E + 255) / 256, 256, 0, stream>>>(ei, xpad, agg1, cnt, E);

  const int tiles  = (N + 15) / 16;
  const int blocks = (tiles + 7) / 8;   // 8 waves (tiles) per 256-thread block
  lin1_kernel<<<blocks, 256, 0, stream>>>(xpad, agg1, cnt, wpad, b1, h, g, N);

  scatter2_kernel<<<(E + 255) / 256, 256, 0, stream>>>(ei, g, aggg, E);

  lin2_kernel<<<blocks, 256, 0, stream>>>(h, aggg, cnt, wpad, (float*)d_out, N);
}